// TeacherGNN_12627203850283
// MI455X (gfx1250) — compile-verified
//
#include <hip/hip_runtime.h>
#include <hip/hip_bf16.h>

// ---------------- problem constants ----------------
#define NN_   100000   // N_NOTE
#define NB_   10000    // N_BAR
#define E_NN_ 1600000
#define E_NB_ 200000
#define E_BN_ 200000
#define HDIM  128

typedef float v2f __attribute__((ext_vector_type(2)));
typedef float v8f __attribute__((ext_vector_type(8)));

// =====================================================================
// WMMA fp32 GEMM:  Y[N x 128] = act( (ACCUM? Y : 0) + X[N x K] @ W[K x 128] + bias )
// One wave per 16-row strip; 8 column tiles of 16 => full 128 columns.
// Uses V_WMMA_F32_16X16X4_F32 (full fp32 precision, matches the f32 reference).
// A frag (16x4): lane = M (lane&15), VGPR0/1 = K / K+1, lane-half selects K base (0 or 2).
// B frag (4x16): lane = N (lane&15), VGPR0/1 = K / K+1, lane-half selects K base.
// C/D (16x16): vgpr r holds M = r + 8*(lane>>4), N = lane&15 (per ISA table).
// GUARD=0: N is a multiple of 16 -> no bounds checks, no EXEC divergence in loop.
// =====================================================================
template<int RELU, int ACCUM, int GUARD>
__global__ void gnn_gemm128(const float* __restrict__ X, const float* __restrict__ W,
                            const float* __restrict__ bias, float* __restrict__ Y,
                            int N, int K) {
    int wave = (int)((blockIdx.x * blockDim.x + threadIdx.x) >> 5);
    int lane = (int)(threadIdx.x & 31);
    int row0 = wave * 16;
    if (row0 >= N) return;                 // wave-uniform exit (EXEC stays full)
    int mr   = lane & 15;                  // M for A, N for B/C
    int half = lane >> 4;
    int row  = row0 + mr;
    bool rowOK = GUARD ? (row < N) : true;

    v8f acc[8] = {};                       // zero accumulators

#pragma unroll 2
    for (int kk = 0; kk < K; kk += 4) {
        int ka = kk + 2 * half;
        v2f a;
        if (GUARD) {
            a.x = rowOK ? X[(size_t)row * K + ka    ] : 0.0f;
            a.y = rowOK ? X[(size_t)row * K + ka + 1] : 0.0f;
        } else {
            a = *(const v2f*)(X + (size_t)row * K + ka);   // single b64 load
        }
        const float* Wr0 = W + (size_t)ka * HDIM;
        const float* Wr1 = W + (size_t)(ka + 1) * HDIM;
#pragma unroll
        for (int t = 0; t < 8; ++t) {
            v2f b;
            b.x = Wr0[t * 16 + mr];
            b.y = Wr1[t * 16 + mr];
            acc[t] = __builtin_amdgcn_wmma_f32_16x16x4_f32(
                false, a, false, b, (short)0, acc[t], false, false);
        }
    }

#pragma unroll
    for (int t = 0; t < 8; ++t) {
        int n = t * 16 + mr;
        float bv = bias ? bias[n] : 0.0f;
#pragma unroll
        for (int r = 0; r < 8; ++r) {
            int m = row0 + r + 8 * half;
            if (!GUARD || (m < N)) {
                size_t idx = (size_t)m * HDIM + n;
                float v = acc[t][r] + bv;
                if (ACCUM) v += Y[idx];
                if (RELU)  v = v > 0.0f ? v : 0.0f;
                Y[idx] = v;
            }
        }
    }
}

// ---------------- edge scatter-add: one wave per edge (512B row, float4/lane) ----
__global__ void gnn_edge_scatter(const int* __restrict__ src, const int* __restrict__ dst,
                                 const float* __restrict__ X, float* __restrict__ S,
                                 float* __restrict__ cnt, int E) {
    int e = (int)(blockIdx.x * (blockDim.x >> 5) + (threadIdx.x >> 5));
    if (e >= E) return;
    int lane = (int)(threadIdx.x & 31);
    int s = src[e], d = dst[e];
    float4 v = ((const float4*)(X + (size_t)s * HDIM))[lane];
    float* sd = S + (size_t)d * HDIM + lane * 4;
    atomicAdd(sd + 0, v.x);
    atomicAdd(sd + 1, v.y);
    atomicAdd(sd + 2, v.z);
    atomicAdd(sd + 3, v.w);
    if (lane == 0) atomicAdd(cnt + d, 1.0f);
}

// ---------------- per-node pooling scatter: one wave per node ----------------
__global__ void gnn_node_scatter(const float* __restrict__ X, const int* __restrict__ seg,
                                 float* __restrict__ S, float* __restrict__ cnt, int N) {
    int i = (int)(blockIdx.x * (blockDim.x >> 5) + (threadIdx.x >> 5));
    if (i >= N) return;
    int lane = (int)(threadIdx.x & 31);
    int d = seg[i];
    float4 v = ((const float4*)(X + (size_t)i * HDIM))[lane];
    float* sd = S + (size_t)d * HDIM + lane * 4;
    atomicAdd(sd + 0, v.x);
    atomicAdd(sd + 1, v.y);
    atomicAdd(sd + 2, v.z);
    atomicAdd(sd + 3, v.w);
    if (lane == 0) atomicAdd(cnt + d, 1.0f);
}

// ---------------- row mean finalize: S[i,:] /= max(cnt[i],1) ----------------
__global__ void gnn_div_rows(float* __restrict__ S, const float* __restrict__ cnt,
                             long long total) {
    long long i = (long long)blockIdx.x * blockDim.x + threadIdx.x;
    if (i >= total) return;
    float c = cnt[i >> 7];
    S[i] = S[i] / fmaxf(c, 1.0f);
}

// ---------------- wsum = A + B (128x128) ----------------
__global__ void gnn_add_mats(const float* __restrict__ A, const float* __restrict__ B,
                             float* __restrict__ O, int n) {
    int i = (int)(blockIdx.x * blockDim.x + threadIdx.x);
    if (i < n) O[i] = A[i] + B[i];
}

// ---------------- LayerNorm (+optional residual, +optional ReLU), one wave/row ----
template<int RELU>
__global__ void gnn_layernorm(const float* __restrict__ U, const float* __restrict__ Res,
                              const float* __restrict__ g, const float* __restrict__ b,
                              float* __restrict__ Y, int N) {
    int row = (int)(blockIdx.x * (blockDim.x >> 5) + (threadIdx.x >> 5));
    if (row >= N) return;
    int lane = (int)(threadIdx.x & 31);
    float4 v = ((const float4*)(U + (size_t)row * HDIM))[lane];
    if (Res) {
        float4 r = ((const float4*)(Res + (size_t)row * HDIM))[lane];
        v.x += r.x; v.y += r.y; v.z += r.z; v.w += r.w;
    }
    float s = v.x + v.y + v.z + v.w;
#pragma unroll
    for (int o = 16; o > 0; o >>= 1) s += __shfl_xor(s, o, 32);
    float mu = s * (1.0f / 128.0f);
    float dx = v.x - mu, dy = v.y - mu, dz = v.z - mu, dw = v.w - mu;
    float q = dx * dx + dy * dy + dz * dz + dw * dw;
#pragma unroll
    for (int o = 16; o > 0; o >>= 1) q += __shfl_xor(q, o, 32);
    float inv = rsqrtf(q * (1.0f / 128.0f) + 1e-5f);
    float4 gv = ((const float4*)g)[lane];
    float4 bv = ((const float4*)b)[lane];
    float4 out;
    out.x = dx * inv * gv.x + bv.x;
    out.y = dy * inv * gv.y + bv.y;
    out.z = dz * inv * gv.z + bv.z;
    out.w = dw * inv * gv.w + bv.w;
    if (RELU) {
        out.x = fmaxf(out.x, 0.0f); out.y = fmaxf(out.y, 0.0f);
        out.z = fmaxf(out.z, 0.0f); out.w = fmaxf(out.w, 0.0f);
    }
    ((float4*)(Y + (size_t)row * HDIM))[lane] = out;
}

// ---------------- pooled finalize + averaged-graph input (8x128) ----------------
__global__ void gnn_pool_finalize(const float* __restrict__ psn, const float* __restrict__ pcn,
                                  const float* __restrict__ psb, const float* __restrict__ pcb,
                                  float* __restrict__ pooled_note, float* __restrict__ pooled_bar,
                                  float* __restrict__ pavg) {
    int j = (int)(blockIdx.x * blockDim.x + threadIdx.x);
    if (j >= 8 * HDIM) return;
    int r = j >> 7;
    float pn = psn[j] / fmaxf(pcn[r], 1.0f);
    float pb = psb[j] / fmaxf(pcb[r], 1.0f);
    pooled_note[j] = pn;
    pooled_bar[j]  = pb;
    pavg[j] = 0.5f * (pn + pb);
}

// ---------------- score head: out[r] = dot(h[r,:], w[:,0]) + b ----------------
__global__ void gnn_score(const float* __restrict__ h, const float* __restrict__ w,
                          const float* __restrict__ b, float* __restrict__ out) {
    int row = (int)(threadIdx.x >> 5);     // 8 waves in one 256-thread block
    int lane = (int)(threadIdx.x & 31);
    float4 v = ((const float4*)(h + (size_t)row * HDIM))[lane];
    float4 ww = ((const float4*)w)[lane];
    float s = v.x * ww.x + v.y * ww.y + v.z * ww.z + v.w * ww.w;
#pragma unroll
    for (int o = 16; o > 0; o >>= 1) s += __shfl_xor(s, o, 32);
    if (lane == 0) out[row] = s + b[0];
}

// =====================================================================
// host orchestration
// =====================================================================
static inline int gemm_blocks(int N) { int waves = (N + 15) / 16; return (waves + 7) / 8; }

extern "C" void kernel_launch(void* const* d_in, const int* in_sizes, int n_in,
                              void* d_out, int out_size, void* d_ws, size_t ws_size,
                              hipStream_t stream) {
    (void)in_sizes; (void)n_in; (void)out_size; (void)ws_size;

    // ---- inputs (setup_inputs dict insertion order, params flattened depth-first) ----
    const float* x_note = (const float*)d_in[0];
    const float* x_bar  = (const float*)d_in[1];
    const int* e_nn = (const int*)d_in[2];   // [2, E_NN]: src then dst
    const int* e_nb = (const int*)d_in[3];
    const int* e_bn = (const int*)d_in[4];
    const int* batch_note = (const int*)d_in[5];
    const int* batch_bar  = (const int*)d_in[6];
    const float* P[64];
    for (int i = 0; i < 61; ++i) P[i] = (const float*)d_in[7 + i];
    // P[0..5]  enc_note: w1,b1,w2,b2,g,b
    // P[6..11] enc_bar
    // P[12 + 13*l + k] layers: nn(wl,bl,wr) nb(wl,bl,wr) bn(wl,bl,wr) ln_note(g,b) ln_bar(g,b)
    // P[51,52] pool wp,bp ; P[53,54] recon_note ; P[55,56] recon_bar ; P[57..60] score

    // ---- d_out layout (floats) ----
    float* out = (float*)d_out;
    float* h_note     = out;                                   // 100000x128
    float* h_bar      = h_note + (size_t)NN_ * HDIM;           // 10000x128
    float* graph_o    = h_bar + (size_t)NB_ * HDIM;            // 8x128
    float* score_o    = graph_o + 8 * HDIM;                    // 8x1
    float* recon_note = score_o + 8;                           // 100000x128
    float* recon_bar  = recon_note + (size_t)NN_ * HDIM;       // 10000x128
    float* pooled_n_o = recon_bar + (size_t)NB_ * HDIM;        // 8x128
    float* pooled_b_o = pooled_n_o + 8 * HDIM;                 // 8x128

    // ---- workspace layout (floats) ----
    float* ws = (float*)d_ws;
    float* M    = ws;                                  // 100000x128  (agg scratch)
    float* UP   = M   + (size_t)NN_ * HDIM;            // 100000x128  (note update / enc hidden)
    float* UPB  = UP  + (size_t)NN_ * HDIM;            // 10000x128   (bar update)
    float* CNT  = UPB + (size_t)NB_ * HDIM;            // 100000
    float* WSUM = CNT + NN_;                           // 128x128
    float* PAVG = WSUM + HDIM * HDIM;                  // 8x128
    float* SCH  = PAVG + 8 * HDIM;                     // 8x128
    float* PSN  = SCH + 8 * HDIM;                      // 8x128
    float* PSB  = PSN + 8 * HDIM;                      // 8x128
    float* PCN  = PSB + 8 * HDIM;                      // 8
    float* PCB  = PCN + 8;                             // 8

    const int TB = 256;
    const int wavesPerBlk = TB / 32;

    // ================= encoders =================
    // note: UP = relu(x @ w1 + b1); M = UP @ w2 + b2; h_note = LN(M)
    gnn_gemm128<1,0,0><<<gemm_blocks(NN_), TB, 0, stream>>>(x_note, P[0], P[1], UP, NN_, 64);
    gnn_gemm128<0,0,0><<<gemm_blocks(NN_), TB, 0, stream>>>(UP, P[2], P[3], M, NN_, HDIM);
    gnn_layernorm<0><<<(NN_ + wavesPerBlk - 1) / wavesPerBlk, TB, 0, stream>>>(
        M, nullptr, P[4], P[5], h_note, NN_);
    // bar
    gnn_gemm128<1,0,0><<<gemm_blocks(NB_), TB, 0, stream>>>(x_bar, P[6], P[7], UPB, NB_, 32);
    gnn_gemm128<0,0,0><<<gemm_blocks(NB_), TB, 0, stream>>>(UPB, P[8], P[9], M, NB_, HDIM);
    gnn_layernorm<0><<<(NB_ + wavesPerBlk - 1) / wavesPerBlk, TB, 0, stream>>>(
        M, nullptr, P[10], P[11], h_bar, NB_);

    // ================= 3 hetero-SAGE layers =================
    for (int l = 0; l < 3; ++l) {
        const float* wl_nn = P[12 + 13 * l + 0];
        const float* bl_nn = P[12 + 13 * l + 1];
        const float* wr_nn = P[12 + 13 * l + 2];
        const float* wl_nb = P[12 + 13 * l + 3];
        const float* bl_nb = P[12 + 13 * l + 4];
        const float* wr_nb = P[12 + 13 * l + 5];
        const float* wl_bn = P[12 + 13 * l + 6];
        const float* bl_bn = P[12 + 13 * l + 7];
        const float* wr_bn = P[12 + 13 * l + 8];
        const float* g_n = P[12 + 13 * l + 9],  *b_n = P[12 + 13 * l + 10];
        const float* g_b = P[12 + 13 * l + 11], *b_b = P[12 + 13 * l + 12];

        // ---- note <- nn edges: UP = mean_nn @ wl + bl
        hipMemsetAsync(M,   0, (size_t)NN_ * HDIM * sizeof(float), stream);
        hipMemsetAsync(CNT, 0, (size_t)NN_ * sizeof(float), stream);
        gnn_edge_scatter<<<(E_NN_ + wavesPerBlk - 1) / wavesPerBlk, TB, 0, stream>>>(
            e_nn, e_nn + E_NN_, h_note, M, CNT, E_NN_);
        gnn_div_rows<<<(int)(((long long)NN_ * HDIM + TB - 1) / TB), TB, 0, stream>>>(
            M, CNT, (long long)NN_ * HDIM);
        gnn_gemm128<0,0,0><<<gemm_blocks(NN_), TB, 0, stream>>>(M, wl_nn, bl_nn, UP, NN_, HDIM);

        // ---- note <- bn edges: UP += mean_bn @ wl + bl
        hipMemsetAsync(M,   0, (size_t)NN_ * HDIM * sizeof(float), stream);
        hipMemsetAsync(CNT, 0, (size_t)NN_ * sizeof(float), stream);
        gnn_edge_scatter<<<(E_BN_ + wavesPerBlk - 1) / wavesPerBlk, TB, 0, stream>>>(
            e_bn, e_bn + E_BN_, h_bar, M, CNT, E_BN_);
        gnn_div_rows<<<(int)(((long long)NN_ * HDIM + TB - 1) / TB), TB, 0, stream>>>(
            M, CNT, (long long)NN_ * HDIM);
        gnn_gemm128<0,1,0><<<gemm_blocks(NN_), TB, 0, stream>>>(M, wl_bn, bl_bn, UP, NN_, HDIM);

        // ---- note self terms: UP += h_note @ (wr_nn + wr_bn)
        gnn_add_mats<<<(HDIM * HDIM + TB - 1) / TB, TB, 0, stream>>>(wr_nn, wr_bn, WSUM, HDIM * HDIM);
        gnn_gemm128<0,1,0><<<gemm_blocks(NN_), TB, 0, stream>>>(h_note, WSUM, nullptr, UP, NN_, HDIM);

        // ---- bar <- nb edges (uses pre-update h_note): UPB = mean_nb @ wl + bl + h_bar @ wr
        hipMemsetAsync(M,   0, (size_t)NB_ * HDIM * sizeof(float), stream);
        hipMemsetAsync(CNT, 0, (size_t)NB_ * sizeof(float), stream);
        gnn_edge_scatter<<<(E_NB_ + wavesPerBlk - 1) / wavesPerBlk, TB, 0, stream>>>(
            e_nb, e_nb + E_NB_, h_note, M, CNT, E_NB_);
        gnn_div_rows<<<(int)(((long long)NB_ * HDIM + TB - 1) / TB), TB, 0, stream>>>(
            M, CNT, (long long)NB_ * HDIM);
        gnn_gemm128<0,0,0><<<gemm_blocks(NB_), TB, 0, stream>>>(M, wl_nb, bl_nb, UPB, NB_, HDIM);
        gnn_gemm128<0,1,0><<<gemm_blocks(NB_), TB, 0, stream>>>(h_bar, wr_nb, nullptr, UPB, NB_, HDIM);

        // ---- residual + LN + relu (in-place update of h)
        gnn_layernorm<1><<<(NN_ + wavesPerBlk - 1) / wavesPerBlk, TB, 0, stream>>>(
            UP, h_note, g_n, b_n, h_note, NN_);
        gnn_layernorm<1><<<(NB_ + wavesPerBlk - 1) / wavesPerBlk, TB, 0, stream>>>(
            UPB, h_bar, g_b, b_b, h_bar, NB_);
    }

    // ================= pooling + heads =================
    hipMemsetAsync(PSN, 0, (2 * 8 * HDIM + 16) * sizeof(float), stream);  // PSN, PSB, PCN, PCB
    gnn_node_scatter<<<(NN_ + wavesPerBlk - 1) / wavesPerBlk, TB, 0, stream>>>(
        h_note, batch_note, PSN, PCN, NN_);
    gnn_node_scatter<<<(NB_ + wavesPerBlk - 1) / wavesPerBlk, TB, 0, stream>>>(
        h_bar, batch_bar, PSB, PCB, NB_);
    gnn_pool_finalize<<<(8 * HDIM + TB - 1) / TB, TB, 0, stream>>>(
        PSN, PCN, PSB, PCB, pooled_n_o, pooled_b_o, PAVG);

    // graph = PAVG @ wp + bp   (N=8 -> guarded variant)
    gnn_gemm128<0,0,1><<<gemm_blocks(8), TB, 0, stream>>>(PAVG, P[51], P[52], graph_o, 8, HDIM);
    // score = relu(graph @ w1 + b1) @ w2 + b2
    gnn_gemm128<1,0,1><<<gemm_blocks(8), TB, 0, stream>>>(graph_o, P[57], P[58], SCH, 8, HDIM);
    gnn_score<<<1, TB, 0, stream>>>(SCH, P[59], P[60], score_o);

    // reconstructions
    gnn_gemm128<0,0,0><<<gemm_blocks(NN_), TB, 0, stream>>>(h_note, P[53], P[54], recon_note, NN_, HDIM);
    gnn_gemm128<0,0,0><<<gemm_blocks(NB_), TB, 0, stream>>>(h_bar, P[55], P[56], recon_bar, NB_, HDIM);
}